// SegmentedSmoothing_15831249453235
// MI455X (gfx1250) — compile-verified
//
#include <hip/hip_runtime.h>
#include <stdint.h>

#define H_DIM 720
#define W_DIM 1440
#define BWID  10
#define TW    64
#define TH    32
#define HT    (TH + 4)      // 36 row slots (output rows + halo)
#define WTMAX (TW + 4)      // 68 max loaded columns

typedef unsigned int u32;
typedef __attribute__((ext_vector_type(4))) u32 u32x4_t;
typedef __attribute__((ext_vector_type(8))) int i32x8_t;
typedef __attribute__((ext_vector_type(4))) int i32x4_t;

// Build the common 2-D D# (descriptor) for a tile of `th` rows x `tw` fp32
// elements with global row stride `stride` elems, LDS side packed at pitch
// tw*4 bytes. Used for both TENSOR_LOAD_TO_LDS and TENSOR_STORE_FROM_LDS.
__device__ __forceinline__ void tdm_make_desc(u32x4_t& g0, i32x8_t& g1,
                                              u32 lds_off, unsigned long long gaddr,
                                              u32 tw, u32 th, u32 stride) {
  // ---- group 0 (128b): count=1 | lds_addr | global_addr[56:0] | type=2
  g0.x = 1u;                                               // count=1, user mode
  g0.y = lds_off;                                          // LDS byte address
  g0.z = (u32)(gaddr & 0xFFFFFFFFull);                     // global_addr[31:0]
  g0.w = ((u32)(gaddr >> 32) & 0x01FFFFFFu) | 0x80000000u; // addr[56:32] | type=2
  // ---- group 1 (256b)
  g1[0] = (int)(2u << 16);                                 // wg_mask=0, data_size=4B
  g1[1] = (int)((tw & 0xFFFFu) << 16);                     // tensor_dim0[15:0] @63:48
  g1[2] = (int)((tw >> 16) | ((th & 0xFFFFu) << 16));      // td0[31:16] | td1[15:0]
  g1[3] = (int)((th >> 16) | ((tw & 0xFFFFu) << 16));      // td1[31:16] | tile_dim0
  g1[4] = (int)(th & 0xFFFFu);                             // tile_dim1 | tile_dim2=0
  g1[5] = (int)stride;                                     // tensor_dim0_stride[31:0]
  g1[6] = 0;                                               // stride0 hi | stride1 lo
  g1[7] = 0;                                               // stride1 hi
}

__device__ __forceinline__ void tdm_load_tile(u32 lds_off, unsigned long long gaddr,
                                              u32 tw, u32 th, u32 stride) {
  u32x4_t g0; i32x8_t g1;
  tdm_make_desc(g0, g1, lds_off, gaddr, tw, th, stride);
  i32x4_t gz; gz[0] = 0; gz[1] = 0; gz[2] = 0; gz[3] = 0;  // groups 2/3 unused (2D)
#if defined(__clang_major__) && (__clang_major__ >= 23)
  i32x8_t gz8;
  gz8[0]=0; gz8[1]=0; gz8[2]=0; gz8[3]=0; gz8[4]=0; gz8[5]=0; gz8[6]=0; gz8[7]=0;
  __builtin_amdgcn_tensor_load_to_lds(g0, g1, gz, gz, gz8, 0);
#else
  __builtin_amdgcn_tensor_load_to_lds(g0, g1, gz, gz, 0);
#endif
}

__device__ __forceinline__ void tdm_store_tile(u32 lds_off, unsigned long long gaddr,
                                               u32 tw, u32 th, u32 stride) {
  u32x4_t g0; i32x8_t g1;
  tdm_make_desc(g0, g1, lds_off, gaddr, tw, th, stride);
  i32x4_t gz; gz[0] = 0; gz[1] = 0; gz[2] = 0; gz[3] = 0;
#if defined(__clang_major__) && (__clang_major__ >= 23)
  i32x8_t gz8;
  gz8[0]=0; gz8[1]=0; gz8[2]=0; gz8[3]=0; gz8[4]=0; gz8[5]=0; gz8[6]=0; gz8[7]=0;
  __builtin_amdgcn_tensor_store_from_lds(g0, g1, gz, gz, gz8, 0);
#else
  __builtin_amdgcn_tensor_store_from_lds(g0, g1, gz, gz, 0);
#endif
}

__global__ __launch_bounds__(256) void SegmentedSmoothing_kernel(
    const float* __restrict__ x, float* __restrict__ out) {
  __shared__ float xt[HT * WTMAX];   // in: staged tile+halo; later reused as out-stage
  __shared__ float hAll[HT * TW];    // horizontal 5-sum of x
  __shared__ float hInt[HT * TW];    // horizontal 5-sum of x * interior_mask

  const int tid  = threadIdx.x;
  const int col0 = blockIdx.x * TW;
  const int row0 = blockIdx.y * TH;
  const int img  = blockIdx.z;

  // loaded region = (tile + halo) clamped to the image
  const int r0l = max(row0 - 2, 0);
  const int r1l = min(row0 + TH + 2, H_DIM);
  const int c0l = max(col0 - 2, 0);
  const int c1l = min(col0 + TW + 2, W_DIM);
  const int loadH = r1l - r0l;
  const int loadW = c1l - c0l;
  // valid (in-image) output tile extent
  const int validW = min(W_DIM - col0, TW);
  const int validH = min(H_DIM - row0, TH);

  // ---- stage input tile via Tensor Data Mover (wave 0 issues, EXEC-independent)
  if (tid < 32) {
    unsigned long long gaddr = (unsigned long long)(uintptr_t)
        (x + ((size_t)img * H_DIM + (size_t)r0l) * W_DIM + c0l);
    tdm_load_tile((u32)(size_t)(&xt[0]), gaddr, (u32)loadW, (u32)loadH, (u32)W_DIM);
    __builtin_amdgcn_s_wait_tensorcnt(0);
  }
  __syncthreads();

  // ---- phase 1: horizontal 5-sums for every loaded row slot
  for (int w = tid; w < HT * TW; w += 256) {
    const int rs = w >> 6;              // row slot 0..35  (TW==64)
    const int jj = w & (TW - 1);
    const int qi = row0 - 2 + rs;       // global row of this slot
    const int j  = col0 + jj;           // global output column
    float sa = 0.f, si = 0.f;
    if (qi >= 0 && qi < H_DIM && j < W_DIM) {
      const int xbase = (qi - r0l) * loadW - c0l;  // xt[xbase + qj]
      #pragma unroll
      for (int d = -2; d <= 2; ++d) {
        const int qj = j + d;
        if (qj >= 0 && qj < W_DIM) {
          const float v = xt[xbase + qj];
          sa += v;
          if (qj >= BWID && qj < W_DIM - BWID) si += v;  // interior column
        }
      }
      if (!(qi >= BWID && qi < H_DIM - BWID)) si = 0.f;  // interior row flag
    }
    hAll[rs * TW + jj] = sa;
    hInt[rs * TW + jj] = si;
  }
  __syncthreads();   // phase-1 done; xt is dead, reuse it as packed output stage

  // ---- phase 2: vertical 5-sums + closed-form counts + select + divide
  {
    const int jj = tid & (TW - 1);
    const int rg = tid >> 6;            // 4 row groups of 8 rows
    const int j  = col0 + jj;
    if (jj < validW) {
      // column window terms (depend only on j)
      const int ac  = max(j - 2, 0), bc = min(j + 2, W_DIM - 1);
      const int nc  = bc - ac + 1;
      const int clo = max(0, min(bc, BWID - 1) - ac + 1);
      const int chi = max(0, bc - max(ac, W_DIM - BWID) + 1);
      const int nbc = clo + chi;        // boundary columns in window
      const bool colB = (j < BWID) || (j >= W_DIM - BWID);

      #pragma unroll
      for (int k = 0; k < 8; ++k) {
        const int ri = rg * 8 + k;
        const int i  = row0 + ri;
        if (ri >= validH) break;

        float wa = 0.f, wi = 0.f;       // slots ri..ri+4 <=> rows i-2..i+2
        #pragma unroll
        for (int s = 0; s < 5; ++s) {
          wa += hAll[(ri + s) * TW + jj];
          wi += hInt[(ri + s) * TW + jj];
        }

        const int ar  = max(i - 2, 0), br = min(i + 2, H_DIM - 1);
        const int nr  = br - ar + 1;
        const int rlo = max(0, min(br, BWID - 1) - ar + 1);
        const int rhi = max(0, br - max(ar, H_DIM - BWID) + 1);
        const int nbr = rlo + rhi;      // boundary rows in window

        const int  cnt_i = (nr - nbr) * (nc - nbc);
        const bool cB    = colB || (i < BWID) || (i >= H_DIM - BWID);
        const float num  = cB ? (wa - wi) : wi;           // num_b = S - S_i
        const int   cnt  = cB ? (nr * nc - cnt_i) : cnt_i;

        xt[ri * validW + jj] = num / (float)cnt;          // packed at pitch validW
      }
    }
  }
  __syncthreads();   // all results staged in LDS

  // ---- drain output tile via Tensor Data Mover (LDS -> global)
  if (tid < 32) {
    unsigned long long gaddr = (unsigned long long)(uintptr_t)
        (out + ((size_t)img * H_DIM + (size_t)row0) * W_DIM + col0);
    tdm_store_tile((u32)(size_t)(&xt[0]), gaddr, (u32)validW, (u32)validH, (u32)W_DIM);
    __builtin_amdgcn_s_wait_tensorcnt(0);
  }
}

extern "C" void kernel_launch(void* const* d_in, const int* in_sizes, int n_in,
                              void* d_out, int out_size, void* d_ws, size_t ws_size,
                              hipStream_t stream) {
  const float* x = (const float*)d_in[0];
  float* out = (float*)d_out;
  const int planes = in_sizes[0] / (H_DIM * W_DIM);   // 4*16 = 64
  dim3 grid((W_DIM + TW - 1) / TW, (H_DIM + TH - 1) / TH, planes);
  dim3 block(256);
  SegmentedSmoothing_kernel<<<grid, block, 0, stream>>>(x, out);
}